// WaveletPacketTransform_74715251081585
// MI455X (gfx1250) — compile-verified
//
#include <hip/hip_runtime.h>

typedef __attribute__((ext_vector_type(2))) float v2f;
typedef __attribute__((ext_vector_type(4))) float v4f;
typedef __attribute__((ext_vector_type(8))) float v8f;
typedef __attribute__((ext_vector_type(4))) int   v4i;

#define AS1 __attribute__((address_space(1)))
#define AS3 __attribute__((address_space(3)))

// Only probe device builtins during the device pass; the host pass has no
// amdgcn builtins declared and would trip these checks spuriously.
#if defined(__HIP_DEVICE_COMPILE__)
#if !__has_builtin(__builtin_amdgcn_wmma_f32_16x16x4_f32)
#error "gfx1250: __builtin_amdgcn_wmma_f32_16x16x4_f32 not declared"
#endif
#if __has_builtin(__builtin_amdgcn_global_load_async_to_lds_b128)
#define HAVE_ASYNC_LDS 1
#else
#define HAVE_ASYNC_LDS 0
#endif
#if __has_builtin(__builtin_amdgcn_s_wait_asynccnt)
#define WAIT_ASYNC(n) __builtin_amdgcn_s_wait_asynccnt(n)
#else
#define WAIT_ASYNC(n) asm volatile("s_wait_asynccnt %0" ::"i"(n))
#endif
#else
#define HAVE_ASYNC_LDS 0
#define WAIT_ASYNC(n)
#endif

// Level-3 Haar wavelet-packet matrix, freq (Gray-code) ordered rows.
// out[p] uses natural band b = p ^ (p>>1); sign for sample j is
// (-1)^( b2&j0 ^ b1&j1 ^ b0&j2 ), scale 2^-1.5.
__device__ __forceinline__ float haarT(int p, int j) {
  const int b = p ^ (p >> 1);
  const int s = ((b >> 2) & j & 1) ^ ((b >> 1) & (j >> 1) & 1) ^ (b & (j >> 2) & 1);
  return s ? -0.35355339059327373f : 0.35355339059327373f;
}

// 16x16 A = blockdiag(T8, T8)
__device__ __forceinline__ float aVal(int m, int k) {
  return (((m ^ k) & 8) == 0) ? haarT(m & 7, k & 7) : 0.0f;
}

// Stage one 2 KiB tile-pair (512 floats) into this wave's LDS buffer.
__device__ __forceinline__ void stage_pair(const float* x, long long gbase,
                                           float* dst, int lane) {
#if HAVE_ASYNC_LDS
  AS1 v4i* g = (AS1 v4i*)(const_cast<float*>(x) + gbase + (lane << 2));
  AS3 v4i* l = (AS3 v4i*)(dst + (lane << 2));
  __builtin_amdgcn_global_load_async_to_lds_b128(g, l, 0, 0);
  __builtin_amdgcn_global_load_async_to_lds_b128(g + 32, l + 32, 0, 0);   // +512 B
  __builtin_amdgcn_global_load_async_to_lds_b128(g + 64, l + 64, 0, 0);   // +1 KiB
  __builtin_amdgcn_global_load_async_to_lds_b128(g + 96, l + 96, 0, 0);   // +1.5 KiB
#else
  *(v4f*)(dst + (lane << 2))       = *(const v4f*)(x + gbase + (lane << 2));
  *(v4f*)(dst + (lane << 2) + 128) = *(const v4f*)(x + gbase + (lane << 2) + 128);
  *(v4f*)(dst + (lane << 2) + 256) = *(const v4f*)(x + gbase + (lane << 2) + 256);
  *(v4f*)(dst + (lane << 2) + 384) = *(const v4f*)(x + gbase + (lane << 2) + 384);
#endif
}

// One 16x16 WMMA chain: 32 groups (256 floats) -> 8 bands x 32 coefficients.
__device__ __forceinline__ v8f wpt_chain(const float* buf, int bbase,
                                         v2f a0, v2f a1, v2f a2, v2f a3) {
  const v2f b0 = *(const v2f*)(buf + bbase);
  const v2f b1 = *(const v2f*)(buf + bbase + 4);
  const v2f b2 = *(const v2f*)(buf + bbase + 128);
  const v2f b3 = *(const v2f*)(buf + bbase + 132);
  v8f c = {};
  c = __builtin_amdgcn_wmma_f32_16x16x4_f32(false, a0, false, b0, (short)0, c, false, false);
  c = __builtin_amdgcn_wmma_f32_16x16x4_f32(false, a1, false, b1, (short)0, c, false, false);
  c = __builtin_amdgcn_wmma_f32_16x16x4_f32(false, a2, false, b2, (short)0, c, false, false);
  c = __builtin_amdgcn_wmma_f32_16x16x4_f32(false, a3, false, b3, (short)0, c, false, false);
  return c;
}

__global__ __launch_bounds__(256) void haar_wpt3_wmma(const float* __restrict__ x,
                                                      float* __restrict__ out,
                                                      long long ntp) {
  __shared__ __align__(16) float lds[8 * 1024];  // 8 waves * (2 buffers * 512 floats)
  const int lane = threadIdx.x & 31;
  const int wib  = threadIdx.x >> 5;
  float* wbuf = &lds[wib << 10];

  // A operand: ISA 32-bit A layout — lanes 0-15 carry row M=lane with (K0,K1)
  // in the 2 VGPRs, lanes 16-31 carry row M=lane-16 with (K2,K3).
  const int m  = lane & 15;
  const int kb = (lane < 16) ? 0 : 2;
  v2f a0, a1, a2, a3;
  a0.x = aVal(m, kb + 0);  a0.y = aVal(m, kb + 1);
  a1.x = aVal(m, kb + 4);  a1.y = aVal(m, kb + 5);
  a2.x = aVal(m, kb + 8);  a2.y = aVal(m, kb + 9);
  a3.x = aVal(m, kb + 12); a3.y = aVal(m, kb + 13);

  // B operand gather base (float index into a 256-float tile):
  // lanes 0-15 need K={4c,4c+1} of column lane  -> tile[8*lane + ...]
  // lanes 16-31 need K={4c+2,4c+3} of col lane' -> tile[8*lane' + 2 + ...]
  const int bbase = (lane < 16) ? (lane << 3) : (((lane - 16) << 3) + 2);

  const long long wglobal = (long long)((blockIdx.x * blockDim.x + threadIdx.x) >> 5);
  const long long nwaves  = (long long)((gridDim.x * blockDim.x) >> 5);

  long long tp  = wglobal;   // tile-pair index: 512 floats (64 groups) each
  int       par = 0;
  if (tp < ntp) stage_pair(x, tp << 9, wbuf, lane);  // prologue -> buffer 0

  for (; tp < ntp; tp += nwaves) {
    // Pipeline: issue next pair's loads, then wait only for the current pair.
    const long long tpn = tp + nwaves;
    if (tpn < ntp) {
      stage_pair(x, tpn << 9, wbuf + ((par ^ 1) << 9), lane);
      WAIT_ASYNC(4);  // async loads complete in order: current 4 done, next 4 in flight
    } else {
      WAIT_ASYNC(0);
    }
    asm volatile("" ::: "memory");  // keep LDS reads below the async wait

    const float* cbuf = wbuf + (par << 9);
    const v8f c0 = wpt_chain(cbuf, bbase, a0, a1, a2, a3);        // groups g0..g0+31
    const v8f c1 = wpt_chain(cbuf + 256, bbase, a0, a1, a2, a3);  // groups g0+32..g0+63

    // VGPR r of each accumulator across all 32 lanes == band r of 32 consecutive
    // groups (lanes<16: D[r][n]; lanes>=16: D[r+8][n-16] — both land at g0+lane).
    const long long gbase = tp << 9;             // 512 floats per pair
    const long long row   = gbase >> 16;         // 65536 floats per (batch,ch) row
    const int       g0row = (int)((gbase & 65535) >> 3);
    float* ob = out + (row << 16) + g0row + lane;  // out row block = 8*8192 floats
#pragma unroll
    for (int r = 0; r < 8; ++r) {
      __builtin_nontemporal_store(c0[r], ob + r * 8192);
      __builtin_nontemporal_store(c1[r], ob + r * 8192 + 32);
    }
    par ^= 1;
  }
}

extern "C" void kernel_launch(void* const* d_in, const int* in_sizes, int n_in,
                              void* d_out, int out_size, void* d_ws, size_t ws_size,
                              hipStream_t stream) {
  (void)n_in; (void)d_ws; (void)ws_size; (void)out_size;
  const float* x = (const float*)d_in[0];
  float* out = (float*)d_out;
  const long long n = (long long)in_sizes[0];   // 64*16*65536 = 2^26
  const long long ntp = n >> 9;                 // 512 floats per tile-pair
  int grid = 2048;                              // 16384 waves -> 8 pairs/wave
  if ((long long)grid * 8 > ntp) grid = (int)((ntp + 7) / 8);
  if (grid < 1) grid = 1;
  haar_wpt3_wmma<<<grid, 256, 0, stream>>>(x, out, ntp);
}